// AMM_66829691126233
// MI455X (gfx1250) — compile-verified
//
#include <hip/hip_runtime.h>
#include <hip/hip_bf16.h>

typedef __attribute__((ext_vector_type(2))) float v2f;
typedef __attribute__((ext_vector_type(8))) float v8f;

#define HW     4096
#define CFM    256
#define CREL   136
#define CTOT   392      // CFM + CREL
#define NPARTS 3
#define LSTR   394      // padded LDS row stride in floats (392 + 2)

__device__ __forceinline__ v8f vzero8() {
  v8f v;
  #pragma unroll
  for (int i = 0; i < 8; ++i) v[i] = 0.0f;
  return v;
}

// ---------------------------------------------------------------------------
// Kernel 1: old_gamma/old_beta (1x1 convs) -> masked g3/b3; zero gamma/beta.
// ---------------------------------------------------------------------------
__global__ void prep_kernel(const float* __restrict__ fm_ref,
                            const float* __restrict__ mask_ref,
                            const float* __restrict__ w_lambda,
                            const float* __restrict__ w_beta,
                            float* __restrict__ g3, float* __restrict__ b3,
                            float* __restrict__ gamma, float* __restrict__ beta) {
  const int p = blockIdx.x * blockDim.x + threadIdx.x;
  if (p >= HW) return;
  float og = 0.0f, ob = 0.0f;
  #pragma unroll 4
  for (int c = 0; c < CFM; ++c) {
    const float v = fm_ref[c * HW + p];
    og = fmaf(v, w_lambda[c], og);
    ob = fmaf(v, w_beta[c], ob);
  }
  gamma[p] = 0.0f;   // accumulators for attn kernel (graph-replay safe)
  beta[p]  = 0.0f;
  #pragma unroll
  for (int b = 0; b < NPARTS; ++b) {
    const float m = mask_ref[b * HW + p];
    g3[b * HW + p] = og * m;
    b3[b * HW + p] = ob * m;
  }
}

// ---------------------------------------------------------------------------
// Vectorized stage of a 64-column block of theta/phi (computed on the fly)
// into the transposed fragment layout dst[col][c] with padded stride LSTR.
// One global_load_b128 (+ mask b128) per 4 columns; 4 ds_store_b32.
// ---------------------------------------------------------------------------
__device__ __forceinline__ void stage_tile(float* __restrict__ dst,
                                           const float* __restrict__ fm,
                                           const float* __restrict__ mask,
                                           const float* __restrict__ rel,
                                           int b, int col0, int tid) {
  for (int idx = tid; idx < CTOT * 16; idx += 128) {
    const int c = idx >> 4;          // channel 0..391
    const int q = idx & 15;          // column quad within the 64-wide block
    const int m = col0 + 4 * q;
    float4 v;
    if (c < CFM) {
      const float4 f  = *(const float4*)(fm + c * HW + m);
      const float4 mk = *(const float4*)(mask + b * HW + m);
      v.x = f.x * mk.x * 0.01f;
      v.y = f.y * mk.y * 0.01f;
      v.z = f.z * mk.z * 0.01f;
      v.w = f.w * mk.w * 0.01f;
    } else {
      v = *(const float4*)(rel + (b * CREL + (c - CFM)) * HW + m);
    }
    float* d = dst + (4 * q) * LSTR + c;
    d[0]        = v.x;
    d[LSTR]     = v.y;
    d[2 * LSTR] = v.z;
    d[3 * LSTR] = v.w;
  }
}

// ---------------------------------------------------------------------------
// Kernel 2: fused  scores = theta^T * phi  (fp32 WMMA, K=392)  +
//           online masked softmax (x200) + attn*g3 / attn*b3 reductions.
// Grid: (64 n-blocks, 3 parts), 128 threads (4 waves). Wave w owns rows
// n0+16w..n0+16w+15 and streams all 4096 columns in 64-wide blocks.
// LDS: resident theta tile + DOUBLE-BUFFERED phi blocks (302.6 KB of 320 KB).
// ---------------------------------------------------------------------------
__global__ void attn_kernel(const float* __restrict__ fm_src,
                            const float* __restrict__ fm_ref,
                            const float* __restrict__ msrc,
                            const float* __restrict__ mref,
                            const float* __restrict__ rel_s,
                            const float* __restrict__ rel_r,
                            const float* __restrict__ g3,
                            const float* __restrict__ b3,
                            float* __restrict__ gamma,
                            float* __restrict__ beta) {
  extern __shared__ float smem[];
  float* Tlds = smem;                       // theta^T tile: [nn 0..63][c]
  float* Pb0  = smem + 64 * LSTR;           // phi^T block, buffer 0
  float* Pb1  = smem + 2 * 64 * LSTR;       // phi^T block, buffer 1

  const int b    = blockIdx.y;
  const int n0   = blockIdx.x * 64;
  const int tid  = threadIdx.x;
  const int wave = tid >> 5;
  const int lane = tid & 31;
  const int lh   = lane & 15;   // lane within 16-lane half
  const int half = lane >> 4;   // 0 or 1

  // ---- stage theta tile + first phi block ----
  stage_tile(Tlds, fm_src, msrc, rel_s, b, n0, tid);
  stage_tile(Pb0,  fm_ref, mref, rel_r, b, 0,  tid);
  __syncthreads();

  float M[8], D[8], G[8], Bt[8];
  #pragma unroll
  for (int r = 0; r < 8; ++r) { M[r] = -3.0e38f; D[r] = 0.f; G[r] = 0.f; Bt[r] = 0.f; }

  // A fragment source: lane l<16 -> K = k+{0,1}; lane l>=16 -> K = k+{2,3}
  const float* Trow = Tlds + (wave * 16 + lh) * LSTR + 2 * half;

  for (int mb = 0; mb < 64; ++mb) {
    const int m0 = mb * 64;
    const float* P = (mb & 1) ? Pb1 : Pb0;

    // ---- stage NEXT phi block into the other buffer (overlaps compute) ----
    if (mb + 1 < 64) {
      stage_tile((mb & 1) ? Pb0 : Pb1, fm_ref, mref, rel_r, b, m0 + 64, tid);
      if (mb + 2 < 64) {  // gfx1250 global_prefetch hint, two blocks ahead
        __builtin_prefetch(fm_ref + (tid >> 6) * HW + m0 + 128 + (tid & 63), 0, 1);
      }
    }

    v8f acc0 = vzero8(), acc1 = vzero8(), acc2 = vzero8(), acc3 = vzero8();
    const float* P0 = P + ( 0 + lh) * LSTR + 2 * half;
    const float* P1 = P + (16 + lh) * LSTR + 2 * half;
    const float* P2 = P + (32 + lh) * LSTR + 2 * half;
    const float* P3 = P + (48 + lh) * LSTR + 2 * half;

    #pragma unroll 2
    for (int k = 0; k < CTOT; k += 4) {
      const v2f A  = *(const v2f*)(Trow + k);
      const v2f B0 = *(const v2f*)(P0 + k);
      const v2f B1 = *(const v2f*)(P1 + k);
      const v2f B2 = *(const v2f*)(P2 + k);
      const v2f B3 = *(const v2f*)(P3 + k);
      acc0 = __builtin_amdgcn_wmma_f32_16x16x4_f32(false, A, false, B0, (short)0, acc0, false, false);
      acc1 = __builtin_amdgcn_wmma_f32_16x16x4_f32(false, A, false, B1, (short)0, acc1, false, false);
      acc2 = __builtin_amdgcn_wmma_f32_16x16x4_f32(false, A, false, B2, (short)0, acc2, false, false);
      acc3 = __builtin_amdgcn_wmma_f32_16x16x4_f32(false, A, false, B3, (short)0, acc3, false, false);
    }

    // ---- online masked softmax update over these 64 columns ----
    const float gv0 = g3[b * HW + m0 +  0 + lh];
    const float gv1 = g3[b * HW + m0 + 16 + lh];
    const float gv2 = g3[b * HW + m0 + 32 + lh];
    const float gv3 = g3[b * HW + m0 + 48 + lh];
    const float bv0 = b3[b * HW + m0 +  0 + lh];
    const float bv1 = b3[b * HW + m0 + 16 + lh];
    const float bv2 = b3[b * HW + m0 + 32 + lh];
    const float bv3 = b3[b * HW + m0 + 48 + lh];

    #pragma unroll
    for (int r = 0; r < 8; ++r) {
      const float s0 = acc0[r], s1 = acc1[r], s2 = acc2[r], s3 = acc3[r];
      const float t0 = s0 * 200.f, t1 = s1 * 200.f, t2 = s2 * 200.f, t3 = s3 * 200.f;
      float mx = fmaxf(fmaxf(t0, t1), fmaxf(t2, t3));
      mx = fmaxf(mx, __shfl_xor(mx, 1, 16));
      mx = fmaxf(mx, __shfl_xor(mx, 2, 16));
      mx = fmaxf(mx, __shfl_xor(mx, 4, 16));
      mx = fmaxf(mx, __shfl_xor(mx, 8, 16));
      const float newM  = fmaxf(M[r], mx);
      const float scale = __expf(M[r] - newM);
      const float e0 = __expf(t0 - newM);
      const float e1 = __expf(t1 - newM);
      const float e2 = __expf(t2 - newM);
      const float e3 = __expf(t3 - newM);
      float dsum = e0 + e1 + e2 + e3;   // denominator: ALL columns
      // numerators: only columns with score != 0 (reference's sparse mask)
      float gsum = e0 * (s0 != 0.f ? gv0 : 0.f) + e1 * (s1 != 0.f ? gv1 : 0.f)
                 + e2 * (s2 != 0.f ? gv2 : 0.f) + e3 * (s3 != 0.f ? gv3 : 0.f);
      float bsum = e0 * (s0 != 0.f ? bv0 : 0.f) + e1 * (s1 != 0.f ? bv1 : 0.f)
                 + e2 * (s2 != 0.f ? bv2 : 0.f) + e3 * (s3 != 0.f ? bv3 : 0.f);
      dsum += __shfl_xor(dsum, 1, 16);
      dsum += __shfl_xor(dsum, 2, 16);
      dsum += __shfl_xor(dsum, 4, 16);
      dsum += __shfl_xor(dsum, 8, 16);
      gsum += __shfl_xor(gsum, 1, 16);
      gsum += __shfl_xor(gsum, 2, 16);
      gsum += __shfl_xor(gsum, 4, 16);
      gsum += __shfl_xor(gsum, 8, 16);
      bsum += __shfl_xor(bsum, 1, 16);
      bsum += __shfl_xor(bsum, 2, 16);
      bsum += __shfl_xor(bsum, 4, 16);
      bsum += __shfl_xor(bsum, 8, 16);
      D[r]  = D[r]  * scale + dsum;
      G[r]  = G[r]  * scale + gsum;
      Bt[r] = Bt[r] * scale + bsum;
      M[r]  = newM;
    }

    __syncthreads();  // next-block staging visible; this block's reads done
  }

  // C-layout: VGPR r -> row r (lanes 0-15) / row r+8 (lanes 16-31)
  if (lh == 0) {
    #pragma unroll
    for (int r = 0; r < 8; ++r) {
      const int n = n0 + wave * 16 + half * 8 + r;
      atomicAdd(&gamma[n], G[r] / D[r]);
      atomicAdd(&beta[n],  Bt[r] / D[r]);
    }
  }
}

// ---------------------------------------------------------------------------
// Kernel 3: out = fm_src * (1 + gamma) + beta
// ---------------------------------------------------------------------------
__global__ void final_kernel(const float* __restrict__ fm_src,
                             const float* __restrict__ gamma,
                             const float* __restrict__ beta,
                             float* __restrict__ out) {
  const int i = blockIdx.x * blockDim.x + threadIdx.x;
  if (i >= CFM * HW) return;
  const int p = i & (HW - 1);
  out[i] = fmaf(fm_src[i], 1.0f + gamma[p], beta[p]);
}

// ---------------------------------------------------------------------------
extern "C" void kernel_launch(void* const* d_in, const int* in_sizes, int n_in,
                              void* d_out, int out_size, void* d_ws, size_t ws_size,
                              hipStream_t stream) {
  const float* fm_src = (const float*)d_in[0];
  const float* fm_ref = (const float*)d_in[1];
  const float* msrc   = (const float*)d_in[2];
  const float* mref   = (const float*)d_in[3];
  const float* rel_s  = (const float*)d_in[4];
  const float* rel_r  = (const float*)d_in[5];
  const float* w_l    = (const float*)d_in[6];
  const float* w_b    = (const float*)d_in[7];
  float* out = (float*)d_out;

  // workspace layout (floats): g3[3*HW] b3[3*HW] gamma[HW] beta[HW] = 128 KB
  float* g3    = (float*)d_ws;
  float* b3    = g3 + NPARTS * HW;
  float* gamma = b3 + NPARTS * HW;
  float* beta  = gamma + HW;

  prep_kernel<<<HW / 256, 256, 0, stream>>>(fm_ref, mref, w_l, w_b,
                                            g3, b3, gamma, beta);

  // theta tile + 2 phi buffers: 3 * 64 * LSTR * 4 B = 302,592 B of 320 KB WGP LDS
  const size_t smem = (size_t)3 * 64 * LSTR * sizeof(float);
  attn_kernel<<<dim3(64, NPARTS), 128, smem, stream>>>(fm_src, fm_ref, msrc, mref,
                                                       rel_s, rel_r, g3, b3,
                                                       gamma, beta);

  final_kernel<<<(CFM * HW) / 256, 256, 0, stream>>>(fm_src, gamma, beta, out);
}